// PDE_DDPG_actor_54537494724941
// MI455X (gfx1250) — compile-verified
//
#include <hip/hip_runtime.h>

typedef __attribute__((ext_vector_type(2))) float v2f;
typedef __attribute__((ext_vector_type(8))) float v8f;

#define BATCH 4096
#define SDIM  1024
#define KWIN  1019                 // state_dim - 5
#define NTOT  (BATCH * KWIN)       // 4,173,824 outputs
#define NPAIRS (NTOT / 32)         // 130,432 pairs of 16-output tiles

__global__ __launch_bounds__(256) void actor_wmma_kernel(
    const float* __restrict__ s,   // (4096, 1024)
    const float* __restrict__ w1,  // (64, 6) row-major
    const float* __restrict__ b1,  // (64,)
    const float* __restrict__ w3,  // (1, 64)
    const float* __restrict__ b3,  // (1,)
    float* __restrict__ out)       // (NTOT,) flat, k-major
{
    const int lane  = threadIdx.x & 31;
    const int half  = lane >> 4;      // 0: lanes 0-15, 1: lanes 16-31
    const int mrow  = lane & 15;

    // Wave-uniform loop induction (forces SGPR loop + saddr-form memory ops)
    const int wid = __builtin_amdgcn_readfirstlane(
        (int)((blockIdx.x * blockDim.x + threadIdx.x) >> 5));
    const int nW = (int)((gridDim.x * blockDim.x) >> 5);

    // aux = s[0, -3:]
    const float aux0 = s[SDIM - 3];
    const float aux1 = s[SDIM - 2];
    const float aux2 = s[SDIM - 1];
    const float b3s  = b3[0];

    // ---- A fragments (constant): w1 columns 0..3 as 4 M-tiles of 16x4 f32.
    // A 16x4 layout: lane L holds row M=L%16; VGPR v holds K = v + 2*(L/16).
    v2f A0, A1, A2, A3;
    {
        const int f0 = 2 * half;  // K {0,1} for lanes<16, {2,3} for lanes>=16
#define LOADA(t, R) { int m = (t) * 16 + mrow; \
                      R.x = w1[m * 6 + f0]; R.y = w1[m * 6 + f0 + 1]; }
        LOADA(0, A0) LOADA(1, A1) LOADA(2, A2) LOADA(3, A3)
#undef LOADA
    }

    // ---- C-init fragments: c[m] = b1[m] + aux1*w1[m,4] + aux2*w1[m,5]
    // C/D 16x16 layout: VGPR v, lane L -> M = v + 8*(L/16), N = L%16.
    v8f C0, C1, C2, C3;
    v8f W0, W1, W2, W3f;
#define LOADC(t, CR, WR) \
    _Pragma("unroll") \
    for (int v = 0; v < 8; ++v) { \
        int m = (t) * 16 + v + 8 * half; \
        CR[v] = b1[m] + aux1 * w1[m * 6 + 4] + aux2 * w1[m * 6 + 5]; \
        WR[v] = w3[m]; \
    }
    LOADC(0, C0, W0) LOADC(1, C1, W1) LOADC(2, C2, W2) LOADC(3, C3, W3f)
#undef LOADC

#define WMMA(Af, Bf, Cf) \
    __builtin_amdgcn_wmma_f32_16x16x4_f32(false, Af, false, Bf, (short)0, Cf, false, false)

    // Loop-invariant per-lane load offsets (elements)
    const int voff0 = mrow * SDIM + 2 * half;  // K=0 (lanes<16) / K=2 (lanes>=16)
    const int voff1 = mrow * SDIM + 1;         // K=1 (used by lanes<16)

    // Uniform tile-pair base pointer: n0 = p*32 -> row b0 = n0%4096, col k = n0/4096
    auto base_ptr = [&](int pp) -> const float* {
        int n0 = pp << 5;
        return s + (n0 & (BATCH - 1)) * SDIM + (n0 >> 12);
    };

    if (wid >= NPAIRS) return;

    // Software pipeline: preload first B elements
    const float* sb = base_ptr(wid);
    float e00 = sb[voff0];
    float e01 = sb[voff1];
    float e10 = sb[voff0 + 16 * SDIM];
    float e11 = sb[voff1 + 16 * SDIM];

    for (int p = wid; p < NPAIRS; p += nW) {
        // B fragments: feats^T (4x16); lane L: N=L%16, VGPR v: K=v+2*(L/16).
        //   K=0: body[b,k]  K=1: body[b,k+1]  K=2: body[b,k+2]  K=3: aux0
        v2f B0, B1;
        B0.x = e00; B0.y = half ? aux0 : e01;
        B1.x = e10; B1.y = half ? aux0 : e11;

        // 8 independent WMMAs: h^T tiles for both output tiles
        v8f d00 = WMMA(A0, B0, C0);
        v8f d01 = WMMA(A1, B0, C1);
        v8f d02 = WMMA(A2, B0, C2);
        v8f d03 = WMMA(A3, B0, C3);
        v8f d10 = WMMA(A0, B1, C0);
        v8f d11 = WMMA(A1, B1, C1);
        v8f d12 = WMMA(A2, B1, C2);
        v8f d13 = WMMA(A3, B1, C3);

        // Prefetch next iteration's B elements (hides L2 latency behind stage 2)
        int pn = p + nW;
        const float* sbn = base_ptr(pn < NPAIRS ? pn : wid);
        e00 = sbn[voff0];
        e01 = sbn[voff1];
        e10 = sbn[voff0 + 16 * SDIM];
        e11 = sbn[voff1 + 16 * SDIM];

        // Stage 2: 8 independent FMA chains (4 m-tiles x 2 output tiles)
        float a00 = 0.f, a01 = 0.f, a02 = 0.f, a03 = 0.f;
        float a10 = 0.f, a11 = 0.f, a12 = 0.f, a13 = 0.f;
#pragma unroll
        for (int v = 0; v < 8; ++v) {
            a00 += W0[v]  * fmaxf(d00[v], 0.f);
            a01 += W1[v]  * fmaxf(d01[v], 0.f);
            a02 += W2[v]  * fmaxf(d02[v], 0.f);
            a03 += W3f[v] * fmaxf(d03[v], 0.f);
            a10 += W0[v]  * fmaxf(d10[v], 0.f);
            a11 += W1[v]  * fmaxf(d11[v], 0.f);
            a12 += W2[v]  * fmaxf(d12[v], 0.f);
            a13 += W3f[v] * fmaxf(d13[v], 0.f);
        }
        float accA = (a00 + a01) + (a02 + a03);
        float accB = (a10 + a11) + (a12 + a13);

        // combine M-halves (lanes L <-> L+16 hold same output column)
        int swA = __builtin_amdgcn_ds_swizzle(__builtin_bit_cast(int, accA), 0x401f); // SWAPX16
        int swB = __builtin_amdgcn_ds_swizzle(__builtin_bit_cast(int, accB), 0x401f);
        float totA = accA + __builtin_bit_cast(float, swA);
        float totB = accB + __builtin_bit_cast(float, swB);

        // lanes 0-15 -> tile0 column L ; lanes 16-31 -> tile1 column L-16
        float r = (half ? totB : totA) + b3s;
        r = __builtin_amdgcn_fmed3f(r, -1.0f, 1.0f);  // clip

        out[(p << 5) + lane] = r;  // saddr-form, fully coalesced 128B store
    }
#undef WMMA
}

extern "C" void kernel_launch(void* const* d_in, const int* in_sizes, int n_in,
                              void* d_out, int out_size, void* d_ws, size_t ws_size,
                              hipStream_t stream) {
    (void)in_sizes; (void)n_in; (void)out_size; (void)d_ws; (void)ws_size;
    const float* s  = (const float*)d_in[0];
    const float* w1 = (const float*)d_in[1];
    const float* b1 = (const float*)d_in[2];
    const float* w3 = (const float*)d_in[3];
    const float* b3 = (const float*)d_in[4];
    float* out = (float*)d_out;

    actor_wmma_kernel<<<1024, 256, 0, stream>>>(s, w1, b1, w3, b3, out);
}